// GeneEmbedder_61375082659939
// MI455X (gfx1250) — compile-verified
//
#include <hip/hip_runtime.h>
#include <cstdint>

// Problem constants (from the reference: N=256, M=2000, D=128, 4 bases)
#define GENE_N   256
#define GENE_M   2000
#define GENE_D   128
#define NBASE    4

// Tiling
#define TILE_M   16                       // m columns per block
#define CHUNK_N  16                       // n rows per block
#define NTHREADS 256                      // 8 waves (wave32)
#define NWAVES   (NTHREADS / 32)

#define TAB_ROWS   (TILE_M * NBASE)       // 64 table rows staged
#define TAB_FLOATS (TAB_ROWS * GENE_D)    // 8192 floats = 32 KB
#define TAB_F4     (TAB_FLOATS / 4)       // 2048 float4
#define ROWS_PER_BLOCK (CHUNK_N * TILE_M) // 256 output rows per block
#define D4         (GENE_D / 4)           // 32 float4 per row (one per lane)

typedef unsigned int u32x4 __attribute__((ext_vector_type(4)));
typedef int          i32x4 __attribute__((ext_vector_type(4)));
typedef int          i32x8 __attribute__((ext_vector_type(8)));
typedef float        f32x4 __attribute__((ext_vector_type(4)));

// Address-space-qualified vector pointer types for the async-LDS builtin:
// param0 = int4 AS(1)* (global src), param1 = int4 AS(3)* (LDS dst).
typedef __attribute__((address_space(1))) i32x4 gas1_i32x4;
typedef __attribute__((address_space(3))) i32x4 gas3_i32x4;

// ---- pick the CDNA5 staging path available on this toolchain ----
#if __has_builtin(__builtin_amdgcn_global_load_async_to_lds_b128)
  #define USE_ASYNC_LDS 1
#elif __has_builtin(__builtin_amdgcn_tensor_load_to_lds)
  #define USE_TDM 1
#endif

__global__ __launch_bounds__(NTHREADS)
void GeneEmbedder_kernel(const int* __restrict__ gseq,
                         const float* __restrict__ emb,
                         float* __restrict__ out)
{
    __shared__ __align__(16) float tab[TAB_FLOATS];   // 32 KB normalized table slice
    __shared__ int sidx[ROWS_PER_BLOCK];              // 1 KB gene indices

    const int tid  = threadIdx.x;
    const int lane = tid & 31;
    const int wave = tid >> 5;
    const int m0   = blockIdx.x * TILE_M;
    const int n0   = blockIdx.y * CHUNK_N;

    // contiguous 32 KB slice: emb[m0 .. m0+TILE_M) x 4 bases x 128 floats
    const float* slice = emb + (size_t)m0 * (NBASE * GENE_D);

    // ---------------- Stage table slice into LDS ----------------
#if defined(USE_ASYNC_LDS)
    // CDNA5 async global->LDS copy (ASYNCcnt-tracked, no VGPR staging).
    // 8 back-to-back b128 async loads per thread, fully unrolled.
    #pragma unroll
    for (int k = 0; k < TAB_F4 / NTHREADS; ++k) {
        const int i = k * NTHREADS + tid;
        __builtin_amdgcn_global_load_async_to_lds_b128(
            (gas1_i32x4*)(slice + 4 * i),
            (gas3_i32x4*)(tab + 4 * i),
            0, 0);
    }
#elif defined(USE_TDM)
    // One TDM descriptor per block: 1 row x 8192 elems x 4 B = 32 KB.
    if (wave == 0) {
        uint64_t ga  = (uint64_t)(uintptr_t)slice;
        uint32_t lds = (uint32_t)(uintptr_t)(void*)tab;   // low 32 bits = LDS byte offset
        u32x4 g0;
        g0.x = 1u;                                        // count=1, user mode
        g0.y = lds;                                       // lds_addr
        g0.z = (uint32_t)ga;                              // global_addr[31:0]
        g0.w = (uint32_t)((ga >> 32) & 0x01FFFFFFu) | (2u << 30); // addr[56:32] | type=2
        i32x8 g1;
        g1[0] = 0x00020000;                               // data_size = 4 B
        g1[1] = (int)(((unsigned)TAB_FLOATS & 0xFFFFu) << 16); // tensor_dim0[15:0]
        g1[2] = (int)(((unsigned)TAB_FLOATS >> 16) | (1u << 16)); // dim0 hi | tensor_dim1=1
        g1[3] = (int)((unsigned)TAB_FLOATS << 16);        // tile_dim0 = 8192
        g1[4] = 1;                                        // tile_dim1 = 1, tile_dim2 = 0
        g1[5] = TAB_FLOATS;                               // tensor_dim0_stride
        g1[6] = 0;
        g1[7] = 0;
        i32x4 g2 = {0, 0, 0, 0};
        i32x4 g3 = {0, 0, 0, 0};
#if __has_include(<hip/amd_detail/amd_gfx1250_TDM.h>)
        i32x8 g45 = {0, 0, 0, 0, 0, 0, 0, 0};            // 6-arg toolchain variant
        __builtin_amdgcn_tensor_load_to_lds(g0, g1, g2, g3, g45, 0);
#else
        __builtin_amdgcn_tensor_load_to_lds(g0, g1, g2, g3, 0);  // ROCm 7.2 5-arg
#endif
    }
#else
    #pragma unroll
    for (int k = 0; k < TAB_F4 / NTHREADS; ++k) {
        const int i = k * NTHREADS + tid;
        ((f32x4*)tab)[i] = ((const f32x4*)slice)[i];
    }
#endif

    // ---- Overlap: load this block's gene indices while the copy is in flight.
    // Read-once data -> non-temporal load (don't pollute L2).
    {
        const int nl = tid / TILE_M;          // 256 threads == 256 indices
        const int ml = tid % TILE_M;
        sidx[tid] = __builtin_nontemporal_load(
            gseq + (size_t)(n0 + nl) * GENE_M + (m0 + ml));
    }

#if defined(USE_ASYNC_LDS)
  #if __has_builtin(__builtin_amdgcn_s_wait_asynccnt)
    __builtin_amdgcn_s_wait_asynccnt(0);
  #else
    asm volatile("s_wait_asynccnt 0" ::: "memory");
  #endif
#elif defined(USE_TDM)
    if (wave == 0) __builtin_amdgcn_s_wait_tensorcnt((short)0);
#endif
    __syncthreads();

    // ---------------- Normalize the 64 staged rows in place ----------------
    // One wave per row: lane holds f32x4 (4 of the 128 elems); shfl_xor reduce.
    #pragma unroll
    for (int k = 0; k < TAB_ROWS / NWAVES; ++k) {
        const int r = k * NWAVES + wave;
        f32x4 v = ((const f32x4*)tab)[r * D4 + lane];
        float ss = v.x * v.x + v.y * v.y + v.z * v.z + v.w * v.w;
        #pragma unroll
        for (int off = 16; off > 0; off >>= 1)
            ss += __shfl_xor(ss, off, 32);
        const float nrm   = sqrtf(ss);
        const float scale = 1.0f / fmaxf(nrm, 1e-12f);
        v *= scale;
        ((f32x4*)tab)[r * D4 + lane] = v;
    }
    __syncthreads();

    // ---------------- Gather: stream the output ----------------
    // Per n-row: the 8 waves cover ml = wave and wave+8 -> two 4 KB contiguous
    // bursts per row. Output is write-once: non-temporal b128 stores so the
    // 262 MB stream rinses to HBM instead of churning L2.
    f32x4* __restrict__ out4 = (f32x4*)out;
    f32x4* orow = out4 + ((size_t)n0 * GENE_M + m0) * D4;       // row n0, col m0
    const int* srow = sidx;
    for (int nl = 0; nl < CHUNK_N; ++nl) {
        #pragma unroll
        for (int j = 0; j < TILE_M / NWAVES; ++j) {             // 2 iters
            const int ml = j * NWAVES + wave;
            const int g  = srow[ml];                            // LDS broadcast, 0..3
            const f32x4 v = ((const f32x4*)tab)[(ml * NBASE + g) * D4 + lane];
            __builtin_nontemporal_store(v, orow + (size_t)ml * D4 + lane);
        }
        orow += (size_t)GENE_M * D4;    // next n row
        srow += TILE_M;
    }
}

extern "C" void kernel_launch(void* const* d_in, const int* in_sizes, int n_in,
                              void* d_out, int out_size, void* d_ws, size_t ws_size,
                              hipStream_t stream) {
    const int*   gseq = (const int*)d_in[0];    // (256, 2000) int32
    const float* emb  = (const float*)d_in[1];  // (2000, 4, 128) float32
    float*       out  = (float*)d_out;          // (256, 2000, 128) float32
    (void)in_sizes; (void)n_in; (void)out_size; (void)d_ws; (void)ws_size;

    dim3 grid(GENE_M / TILE_M, GENE_N / CHUNK_N);   // 125 x 16 = 2000 blocks
    dim3 block(NTHREADS);                           // 8 wave32 waves
    GeneEmbedder_kernel<<<grid, block, 0, stream>>>(gseq, emb, out);
}